// CascadingFullAttention_26010321944673
// MI455X (gfx1250) — compile-verified
//
#include <hip/hip_runtime.h>
#include <hip/hip_bf16.h>

typedef __attribute__((ext_vector_type(2))) float v2f;
typedef __attribute__((ext_vector_type(8))) float v8f;

#define SINK   4
#define RECENT 512
#define MID_W  512
#define MID_S  2
#define FAR_W  1536
#define FAR_S  4
#define LKEYS  (SINK + RECENT + MID_W + FAR_W + 1)   // 2565
#define BATCH  16
#define HQ     32
#define HKV    8
#define REP    (HQ / HKV)                            // 4
#define DH     128
#define HID    4096
#define MAXCTX 8192
#define QKVLD  (HQ*DH + 2*HKV*DH)                    // 6144

// ---------------------------------------------------------------------------
// Closed-form cascading gather index (matches build_gather_indices padding).
// j in [0, LKEYS-1); j == LKEYS-1 is the new token (handled by caller).
// ---------------------------------------------------------------------------
__device__ __forceinline__ int map_index(int j, int P) {
    const int L = P + 1;
    if (j < SINK) return (j < L) ? j : 0;
    j -= SINK;
    int recent_start = max(SINK, L - RECENT);
    if (j < RECENT) {
        int n = L - recent_start; if (n < 0) n = 0;
        int pad = RECENT - n;
        return (j < pad) ? recent_start : recent_start + (j - pad);
    }
    j -= RECENT;
    const int mid_end = recent_start;
    const int mid_start = max(SINK, mid_end - MID_W * MID_S);
    if (j < MID_W) {
        int n = (mid_end - mid_start + MID_S - 1) / MID_S;
        int pad = MID_W - n;
        return (j < pad) ? mid_start : mid_start + MID_S * (j - pad);
    }
    j -= MID_W;
    const int far_end = mid_start;
    const int far_start = max(SINK, far_end - FAR_W * FAR_S);
    int n = (far_end - far_start + FAR_S - 1) / FAR_S;
    int pad = FAR_W - n;
    return (j < pad) ? far_start : far_start + FAR_S * (j - pad);
}

// ---------------------------------------------------------------------------
// GEMM: C[16 x N] = A[16 x K] * B[K x N], fp32 WMMA 16x16x4.
// One wave per 16-column tile, 8 waves per block (wave32).
//   A frag (16x4): lane l -> M = l&15, K pair = (l>>4)*2   (2 VGPRs)
//   B frag (4x16): lane l -> N = l&15, K pair = (l>>4)*2   (2 VGPRs)
//   C/D (16x16):   vgpr r, lane l -> M = r + 8*(l>>4), N = l&15
// Strength-reduced running pointers keep the hot loop to 3 loads + 1 WMMA.
// ---------------------------------------------------------------------------
__global__ void gemm16_wmma(const float* __restrict__ A,
                            const float* __restrict__ B,
                            float* __restrict__ C,
                            int K, int N, int ldc) {
    const int lane = threadIdx.x & 31;
    const int wave = threadIdx.x >> 5;
    const int col0 = (blockIdx.x * (blockDim.x >> 5) + wave) * 16;
    if (col0 >= N) return;

    const int mrow = lane & 15;          // M for A frag, N for B frag
    const int koff = (lane >> 4) * 2;    // 0 or 2

    v8f acc = {};
    const float* aptr = A + (size_t)mrow * K + koff;
    const float* bptr = B + (size_t)koff * N + col0 + mrow;
    const size_t bstep = (size_t)4 * N;

#pragma unroll 4
    for (int k0 = 0; k0 < K; k0 += 4) {
        v2f a = *(const v2f*)aptr;          // contiguous K pair
        v2f b;
        b.x = bptr[0];
        b.y = bptr[N];
        acc = __builtin_amdgcn_wmma_f32_16x16x4_f32(
                  false, a, false, b, (short)0, acc, false, false);
        aptr += 4;
        bptr += bstep;
    }

    const int mbase = (lane >> 4) * 8;
    const int ncol  = col0 + (lane & 15);
#pragma unroll
    for (int r = 0; r < 8; ++r)
        C[(size_t)(mbase + r) * ldc + ncol] = acc[r];
}

// ---------------------------------------------------------------------------
// Per-(b, head-slot) LayerNorm (+affine for q/k) + RoPE (q/k only), scatter.
// blockIdx.x = b*48 + hh; hh<32: q head, 32..39: k head, 40..47: v head.
// 128 threads = one per channel.
// ---------------------------------------------------------------------------
__global__ void postproc_kernel(const float* __restrict__ raw,
                                const float* __restrict__ rope_cos,
                                const float* __restrict__ rope_sin,
                                const float* __restrict__ q_gamma,
                                const float* __restrict__ q_beta,
                                const float* __restrict__ k_gamma,
                                const float* __restrict__ k_beta,
                                const int*   __restrict__ posp,
                                float* __restrict__ qb,
                                float* __restrict__ knew,
                                float* __restrict__ vnew) {
    __shared__ float red[DH];
    __shared__ float sh[DH];
    const int tid = threadIdx.x;
    const int b   = blockIdx.x / 48;
    const int hh  = blockIdx.x % 48;
    const int P   = posp[0];

    int srcofs;
    if (hh < 32)      srcofs = hh * DH;
    else if (hh < 40) srcofs = HQ*DH + (hh - 32) * DH;
    else              srcofs = HQ*DH + HKV*DH + (hh - 40) * DH;

    const float x = raw[(size_t)b * QKVLD + srcofs + tid];

    red[tid] = x; __syncthreads();
#pragma unroll
    for (int s = 64; s > 0; s >>= 1) { if (tid < s) red[tid] += red[tid + s]; __syncthreads(); }
    const float mu = red[0] * (1.0f / DH);
    __syncthreads();

    const float dx = x - mu;
    red[tid] = dx * dx; __syncthreads();
#pragma unroll
    for (int s = 64; s > 0; s >>= 1) { if (tid < s) red[tid] += red[tid + s]; __syncthreads(); }
    const float var = red[0] * (1.0f / DH);

    float y = dx * rsqrtf(var + 1e-5f);
    if (hh < 32)      y = y * q_gamma[tid] + q_beta[tid];
    else if (hh < 40) y = y * k_gamma[tid] + k_beta[tid];

    float out = y;
    if (hh < 40) {   // RoPE for q and k (block-uniform branch)
        sh[tid] = y; __syncthreads();
        const int t = tid & 63;
        const float c  = rope_cos[(size_t)P * (DH/2) + t];
        const float sn = rope_sin[(size_t)P * (DH/2) + t];
        out = (tid < 64) ? (y * c - sh[tid + 64] * sn)
                         : (y * c + sh[t] * sn);
    }

    if (hh < 32)      qb  [((size_t)b*HQ  + hh)      * DH + tid] = out;
    else if (hh < 40) knew[((size_t)b*HKV + (hh-32)) * DH + tid] = out;
    else              vnew[((size_t)b*HKV + (hh-40)) * DH + tid] = out;
}

// ---------------------------------------------------------------------------
// Attention: one block per (b, hkv) — all REP(=4) GQA query heads computed
// per K/V row read, so each gathered K/V element is loaded from memory once.
// 256 threads = 8 waves.
// Phase 1: wave per key; one float4 K load feeds 4 dot products.
// Phase 2: 4 block softmaxes over logits in LDS.
// Phase 3: dim-parallel V accumulation; one V load feeds 4 accumulators.
// ---------------------------------------------------------------------------
__global__ void attn_kernel(const float* __restrict__ qb,
                            const float* __restrict__ knew,
                            const float* __restrict__ vnew,
                            const float* __restrict__ kcache,
                            const float* __restrict__ vcache,
                            const int*   __restrict__ posp,
                            float* __restrict__ attn) {
    __shared__ float logits[REP][LKEYS];
    __shared__ float red[256];
    const int tid  = threadIdx.x;
    const int lane = tid & 31;
    const int wave = tid >> 5;
    const int b    = blockIdx.x >> 3;     // grid = 16*8
    const int hkv  = blockIdx.x & 7;
    const int P    = posp[0];

    const float scale = 0.08838834764831845f;   // 1/sqrt(128)

    float4 qv[REP];
#pragma unroll
    for (int qi = 0; qi < REP; ++qi)
        qv[qi] = *(const float4*)(qb + ((size_t)(b*HQ + hkv*REP + qi)) * DH + lane * 4);

    const float* kbase    = kcache + ((size_t)(b*HKV + hkv)) * MAXCTX * DH;
    const float* vbase    = vcache + ((size_t)(b*HKV + hkv)) * MAXCTX * DH;
    const float* krow_new = knew + (size_t)(b*HKV + hkv) * DH;
    const float* vrow_new = vnew + (size_t)(b*HKV + hkv) * DH;

    // ---- Phase 1: logits --------------------------------------------------
    for (int j = wave; j < LKEYS; j += 8) {
        const float* krow = (j == LKEYS - 1) ? krow_new
                           : (kbase + (size_t)map_index(j, P) * DH);
        const float4 kv = *(const float4*)(krow + lane * 4);
#pragma unroll
        for (int qi = 0; qi < REP; ++qi) {
            float d = qv[qi].x*kv.x + qv[qi].y*kv.y + qv[qi].z*kv.z + qv[qi].w*kv.w;
#pragma unroll
            for (int off = 16; off > 0; off >>= 1) d += __shfl_xor(d, off, 32);
            if (lane == 0) logits[qi][j] = d * scale;
        }
    }
    __syncthreads();

    // ---- Phase 2: softmax (4 heads) ---------------------------------------
    float pm[REP];
#pragma unroll
    for (int qi = 0; qi < REP; ++qi) pm[qi] = -3.402823e38f;
    for (int j = tid; j < LKEYS; j += 256) {
#pragma unroll
        for (int qi = 0; qi < REP; ++qi) pm[qi] = fmaxf(pm[qi], logits[qi][j]);
    }
    float Mh[REP];
#pragma unroll
    for (int qi = 0; qi < REP; ++qi) {
        red[tid] = pm[qi]; __syncthreads();
        for (int s = 128; s > 0; s >>= 1) { if (tid < s) red[tid] = fmaxf(red[tid], red[tid + s]); __syncthreads(); }
        Mh[qi] = red[0]; __syncthreads();
    }

    float ps[REP] = {0.f, 0.f, 0.f, 0.f};
    for (int j = tid; j < LKEYS; j += 256) {
#pragma unroll
        for (int qi = 0; qi < REP; ++qi) {
            const float e = __expf(logits[qi][j] - Mh[qi]);
            logits[qi][j] = e;
            ps[qi] += e;
        }
    }
    float inv[REP];
#pragma unroll
    for (int qi = 0; qi < REP; ++qi) {
        red[tid] = ps[qi]; __syncthreads();
        for (int s = 128; s > 0; s >>= 1) { if (tid < s) red[tid] += red[tid + s]; __syncthreads(); }
        inv[qi] = 1.0f / red[0]; __syncthreads();
    }

    // ---- Phase 3: V accumulation ------------------------------------------
    const int d    = tid & 127;
    const int half = tid >> 7;
    float acc[REP] = {0.f, 0.f, 0.f, 0.f};
    for (int j = half; j < LKEYS; j += 2) {
        const float* vrow = (j == LKEYS - 1) ? vrow_new
                           : (vbase + (size_t)map_index(j, P) * DH);
        const float v = vrow[d];
#pragma unroll
        for (int qi = 0; qi < REP; ++qi) acc[qi] += logits[qi][j] * v;
    }
#pragma unroll
    for (int qi = 0; qi < REP; ++qi) {
        red[tid] = acc[qi]; __syncthreads();
        if (tid < 128)
            attn[((size_t)(b*HQ + hkv*REP + qi)) * DH + d] = (red[tid] + red[tid + 128]) * inv[qi];
        __syncthreads();
    }
}

// ---------------------------------------------------------------------------
extern "C" void kernel_launch(void* const* d_in, const int* in_sizes, int n_in,
                              void* d_out, int out_size, void* d_ws, size_t ws_size,
                              hipStream_t stream) {
    const float* hidden   = (const float*)d_in[0];
    const float* kcache   = (const float*)d_in[1];
    const float* vcache   = (const float*)d_in[2];
    const int*   pos      = (const int*)  d_in[3];
    const float* rope_cos = (const float*)d_in[4];
    const float* rope_sin = (const float*)d_in[5];
    const float* Wq       = (const float*)d_in[6];
    const float* Wk       = (const float*)d_in[7];
    const float* Wv       = (const float*)d_in[8];
    const float* Wo       = (const float*)d_in[9];
    const float* q_gamma  = (const float*)d_in[10];
    const float* q_beta   = (const float*)d_in[11];
    const float* k_gamma  = (const float*)d_in[12];
    const float* k_beta   = (const float*)d_in[13];
    float* out = (float*)d_out;

    float* ws   = (float*)d_ws;
    float* raw  = ws;                              // 16 x 6144
    float* qb   = raw  + (size_t)BATCH * QKVLD;    // 16 x 32 x 128
    float* knew = qb   + (size_t)BATCH * HQ  * DH; // 16 x 8 x 128
    float* vnew = knew + (size_t)BATCH * HKV * DH; // 16 x 8 x 128
    float* attn = vnew + (size_t)BATCH * HKV * DH; // 16 x 4096

    // QKV projections (WMMA fp32)
    gemm16_wmma<<<dim3((HQ*DH)/128),  256, 0, stream>>>(hidden, Wq, raw,               HID, HQ*DH,  QKVLD);
    gemm16_wmma<<<dim3((HKV*DH)/128), 256, 0, stream>>>(hidden, Wk, raw + HQ*DH,       HID, HKV*DH, QKVLD);
    gemm16_wmma<<<dim3((HKV*DH)/128), 256, 0, stream>>>(hidden, Wv, raw + HQ*DH+HKV*DH,HID, HKV*DH, QKVLD);

    // LayerNorm + RoPE + scatter
    postproc_kernel<<<dim3(BATCH * 48), 128, 0, stream>>>(
        raw, rope_cos, rope_sin, q_gamma, q_beta, k_gamma, k_beta, pos,
        qb, knew, vnew);

    // Cascading-gather attention: one block per (b, hkv), 4 q-heads each
    attn_kernel<<<dim3(BATCH * HKV), 256, 0, stream>>>(
        qb, knew, vnew, kcache, vcache, pos, attn);

    // Output projection (WMMA fp32) straight into d_out
    gemm16_wmma<<<dim3((HQ*DH)/128), 256, 0, stream>>>(attn, Wo, out, HID, HQ*DH, HQ*DH);
}